// MeanShift_20907900796935
// MI455X (gfx1250) — compile-verified
//
#include <hip/hip_runtime.h>

typedef __attribute__((ext_vector_type(2))) float v2f;
typedef __attribute__((ext_vector_type(8))) float v8f;

#define CHAN      32
#define NPTS      65536
#define NSEEDS    512
#define BW        64.0f
#define EPS_T     0.01f
#define MAX_STEPS 20
#define LOG2E     1.4426950408889634f
#define KSC       (LOG2E / BW)      // exp(-d) = exp2(-d*log2e)

#define WAVES   16                 // waves per block (512 threads)
#define NCHUNK  (NPTS / WAVES)     // 4096 points per wave
#define NTILES  (NCHUNK / 16)      // 256 16-wide tiles per wave per step

// ---------------------------------------------------------------------------
// Prologue: Xsq[n] = sum_c x[c][n]^2  (coalesced along n per channel)
// ---------------------------------------------------------------------------
__global__ void xsq_kernel(const float* __restrict__ x, float* __restrict__ xsq) {
    int n = blockIdx.x * blockDim.x + threadIdx.x;
    float s = 0.f;
#pragma unroll
    for (int c = 0; c < CHAN; ++c) {
        float v = x[(size_t)c * NPTS + n];
        s += v * v;
    }
    xsq[n] = s;
}

// ---------------------------------------------------------------------------
// Fused mean-shift: one block owns 16 seeds, runs all 20 iterations.
// Both GEMMs use V_WMMA_F32_16X16X4_F32 (full f32 precision parity).
// ---------------------------------------------------------------------------
__global__ __launch_bounds__(WAVES * 32)
void meanshift_kernel(const float* __restrict__ x,
                      const int*   __restrict__ seed_idx,
                      const float* __restrict__ xsq,
                      float*       __restrict__ out) {
    __shared__ float Ccur[16][CHAN];        // current centroids for this seed tile
    __shared__ float Csq16[16];             // ||C_m||^2
    __shared__ float denomp[WAVES][16];     // per-wave denom partials
    __shared__ float diffp[16];             // per-row L1 shift
    __shared__ float state[2];              // [0]=diff carry, [1]=active flag
    union Smem {                            // staging buffers and reduction
        struct {                            // buffer overlay to fit static LDS
            float xt[WAVES][CHAN][17];      // X tile, [c][n], pad 17 (bank-safe)
            float wt[WAVES][16][17];        // w tile, [m][n], pad 17
        } s;
        struct {
            float nump[WAVES][16][CHAN];    // per-wave num partials
        } r;
    };
    __shared__ Smem u;

    const int t    = threadIdx.x;
    const int lane = t & 31;
    const int wv   = t >> 5;        // wave id 0..15
    const int h    = lane >> 4;     // half of wave (K split for WMMA operands)
    const int ln   = lane & 15;

    // Gather seeds: C0[m][c] = X[seed_idx[block*16+m]][c]
    {
        int m = t >> 5, c = t & 31;
        int s = blockIdx.x * 16 + m;
        int idx = seed_idx[s];
        Ccur[m][c] = x[(size_t)c * NPTS + idx];
    }
    if (t == 0) { state[0] = 10.0f; state[1] = 0.f; }
    __syncthreads();

    float (*xt)[17] = u.s.xt[wv];
    float (*wt)[17] = u.s.wt[wv];
    const int nbase = wv * NCHUNK;

    for (int step = 0; step < MAX_STEPS; ++step) {
        // ||C_m||^2
        if (t < 16) {
            float s = 0.f;
#pragma unroll
            for (int c = 0; c < CHAN; ++c) { float v = Ccur[t][c]; s += v * v; }
            Csq16[t] = s;
        }
        __syncthreads();

        // Preload A operand (C rows, 16x4 f32 layout: K = vgpr + 2*half) and
        // per-slot -k*||C_m||^2 (folded into the exp2 argument base).
        v2f a[8];
#pragma unroll
        for (int k0 = 0; k0 < 8; ++k0) {
            a[k0].x = Ccur[ln][4 * k0 + 2 * h];
            a[k0].y = Ccur[ln][4 * k0 + 2 * h + 1];
        }
        float kcsq[8];
#pragma unroll
        for (int v = 0; v < 8; ++v) kcsq[v] = -KSC * Csq16[v + 8 * h];

        v8f nacc0 = {};                 // num accumulator, chans 0..15
        v8f nacc1 = {};                 // num accumulator, chans 16..31
        float dacc[8];
#pragma unroll
        for (int v = 0; v < 8; ++v) dacc[v] = 0.f;

        for (int tile = 0; tile < NTILES; ++tile) {
            const int nb = nbase + tile * 16;

            // Phase-1 B = X^T chunk (k=chan, col=n): coalesced global loads,
            // same registers re-staged into LDS [c][n] for phase 2.
            v2f b1[8];
#pragma unroll
            for (int k0 = 0; k0 < 8; ++k0) {
                int c0 = 4 * k0 + 2 * h;
                float e0 = x[(size_t)c0 * NPTS + nb + ln];
                float e1 = x[(size_t)(c0 + 1) * NPTS + nb + ln];
                b1[k0].x = e0;
                b1[k0].y = e1;
                xt[c0][ln]     = e0;
                xt[c0 + 1][ln] = e1;
            }
            float kxs = -KSC * xsq[nb + ln];

            // Phase 1: dot = C @ X^T  (8 chained f32 WMMAs over K=32 chans)
            v8f acc = {};
#pragma unroll
            for (int k0 = 0; k0 < 8; ++k0)
                acc = __builtin_amdgcn_wmma_f32_16x16x4_f32(
                    false, a[k0], false, b1[k0], (short)0, acc, false, false);

            // w = exp2(2k*dot - k*|C|^2 - k*|X|^2); arg <= 0 so the raw
            // v_exp_f32 (no libm range fixup) is numerically safe here.
            // D layout: vgpr v, lane -> (m = v + 8h, n = ln). Stage w + denom.
#pragma unroll
            for (int v = 0; v < 8; ++v) {
                float arg  = __builtin_fmaf(acc[v], 2.0f * KSC, kcsq[v] + kxs);
                float wexp = __builtin_amdgcn_exp2f(arg);
                dacc[v] += wexp;
                wt[v + 8 * h][ln] = wexp;
            }

            // Phase 2: num += w @ Xtile  (A = w from LDS, B = x tile from LDS)
#pragma unroll
            for (int k = 0; k < 4; ++k) {
                int kk = 4 * k + 2 * h;
                v2f aw;  aw.x  = wt[ln][kk];       aw.y  = wt[ln][kk + 1];
                v2f bx0; bx0.x = xt[ln][kk];       bx0.y = xt[ln][kk + 1];
                v2f bx1; bx1.x = xt[16 + ln][kk];  bx1.y = xt[16 + ln][kk + 1];
                nacc0 = __builtin_amdgcn_wmma_f32_16x16x4_f32(
                    false, aw, false, bx0, (short)0, nacc0, false, false);
                nacc1 = __builtin_amdgcn_wmma_f32_16x16x4_f32(
                    false, aw, false, bx1, (short)0, nacc1, false, false);
            }
        }

        __syncthreads();   // staging buffers dead -> safe to overlay nump

        // Dump num partials (D layout -> [m][c]) and reduce denom in-wave
#pragma unroll
        for (int v = 0; v < 8; ++v) {
            u.r.nump[wv][v + 8 * h][ln]      = nacc0[v];
            u.r.nump[wv][v + 8 * h][16 + ln] = nacc1[v];
        }
#pragma unroll
        for (int v = 0; v < 8; ++v) {
            float d = dacc[v];
            d += __shfl_xor(d, 1, 32);
            d += __shfl_xor(d, 2, 32);
            d += __shfl_xor(d, 4, 32);
            d += __shfl_xor(d, 8, 32);      // sum over 16-lane half (fixed m)
            if (ln == 0) denomp[wv][v + 8 * h] = d;
        }
        __syncthreads();

        // Final reduction + update: thread t owns (m = t>>5, c = t&31);
        // wave m spans exactly the 32 channels of row m.
        {
            int m = t >> 5, c = t & 31;
            float s = 0.f, dn = 0.f;
            for (int w2 = 0; w2 < WAVES; ++w2) {
                s  += u.r.nump[w2][m][c];
                dn += denomp[w2][m];
            }
            float cnew  = s / dn;
            float adiff = fabsf(cnew - Ccur[m][c]);
            adiff += __shfl_xor(adiff, 1, 32);
            adiff += __shfl_xor(adiff, 2, 32);
            adiff += __shfl_xor(adiff, 4, 32);
            adiff += __shfl_xor(adiff, 8, 32);
            adiff += __shfl_xor(adiff, 16, 32);
            if (lane == 0) diffp[m] = adiff;
            __syncthreads();
            if (t == 0) {
                float nd = diffp[0];
                for (int mm = 1; mm < 16; ++mm) nd = fmaxf(nd, diffp[mm]);
                bool active = state[0] > EPS_T;     // emulate data-dependent while
                state[1] = active ? 1.f : 0.f;
                if (active) state[0] = nd;
            }
            __syncthreads();
            if (state[1] != 0.f) Ccur[m][c] = cnew;
        }
        __syncthreads();
    }

    // Write result: out[s][c]
    {
        int m = t >> 5, c = t & 31;
        out[((size_t)blockIdx.x * 16 + m) * CHAN + c] = Ccur[m][c];
    }
}

// ---------------------------------------------------------------------------
extern "C" void kernel_launch(void* const* d_in, const int* in_sizes, int n_in,
                              void* d_out, int out_size, void* d_ws, size_t ws_size,
                              hipStream_t stream) {
    const float* x        = (const float*)d_in[0];
    const int*   seed_idx = (const int*)d_in[1];
    float*       out      = (float*)d_out;
    float*       xsq      = (float*)d_ws;     // N floats = 256 KB scratch

    xsq_kernel<<<dim3(NPTS / 256), dim3(256), 0, stream>>>(x, xsq);
    meanshift_kernel<<<dim3(NSEEDS / 16), dim3(WAVES * 32), 0, stream>>>(
        x, seed_idx, xsq, out);
}